// JacobianForwardKinematics_49289044689564
// MI455X (gfx1250) — compile-verified
//
#include <hip/hip_runtime.h>
#include <hip/hip_bf16.h>

#define DOF 7

#if (__has_builtin(__builtin_amdgcn_global_load_async_to_lds_b32) && \
     __has_builtin(__builtin_amdgcn_s_wait_asynccnt))
#define HAS_ASYNC_LDS 1
#else
#define HAS_ASYNC_LDS 0
#endif

__device__ __forceinline__ void cross3(const float a[3], const float b[3], float c[3]) {
    c[0] = a[1] * b[2] - a[2] * b[1];
    c[1] = a[2] * b[0] - a[0] * b[2];
    c[2] = a[0] * b[1] - a[1] * b[0];
}

__global__ __launch_bounds__(256) void fk_jac_kernel(
    const float* __restrict__ thetas,
    const float* __restrict__ link_trans,
    const float* __restrict__ link_rotvec,
    float* __restrict__ out, int batch)
{
    __shared__ float sRaw[DOF * 6];   // [0..20] link_trans, [21..41] link_rotvec
    __shared__ float sA[DOF][9];      // fixed per-link rotation matrices (row-major)

    const int tid = threadIdx.x;

    // ---- Stage the 42 shared link parameters into LDS (CDNA5 async copy) ----
#if HAS_ASYNC_LDS
    if (tid < DOF * 3) {
        typedef __attribute__((address_space(1))) int GI;   // global src (param is int*)
        typedef __attribute__((address_space(3))) int LI;   // LDS dst
        __builtin_amdgcn_global_load_async_to_lds_b32(
            (GI*)(link_trans + tid),  (LI*)&sRaw[tid],           0, 0);
        __builtin_amdgcn_global_load_async_to_lds_b32(
            (GI*)(link_rotvec + tid), (LI*)&sRaw[DOF * 3 + tid], 0, 0);
    }
    __builtin_amdgcn_s_wait_asynccnt(0);
#else
    if (tid < DOF * 3) {
        sRaw[tid]           = link_trans[tid];
        sRaw[DOF * 3 + tid] = link_rotvec[tid];
    }
#endif
    __syncthreads();

    // ---- Threads 0..6 build the fixed Rodrigues matrices once per block ----
    if (tid < DOF) {
        const float vx = sRaw[DOF * 3 + tid * 3 + 0];
        const float vy = sRaw[DOF * 3 + tid * 3 + 1];
        const float vz = sRaw[DOF * 3 + tid * 3 + 2];
        const float th  = sqrtf(vx * vx + vy * vy + vz * vz + 1e-12f);
        const float inv = 1.0f / th;
        const float ux = vx * inv, uy = vy * inv, uz = vz * inv;
        const float s = sinf(th), c = cosf(th), oc = 1.0f - c;
        const float uu = ux * ux + uy * uy + uz * uz;   // == v.v/(v.v+eps), ~1
        // R = I + s*K + oc*K^2,  K = skew(u),  K^2 = uu^T - (u.u) I
        sA[tid][0] = 1.0f + oc * (ux * ux - uu);
        sA[tid][1] = -s * uz + oc * (ux * uy);
        sA[tid][2] =  s * uy + oc * (ux * uz);
        sA[tid][3] =  s * uz + oc * (uy * ux);
        sA[tid][4] = 1.0f + oc * (uy * uy - uu);
        sA[tid][5] = -s * ux + oc * (uy * uz);
        sA[tid][6] = -s * uy + oc * (uz * ux);
        sA[tid][7] =  s * ux + oc * (uz * uy);
        sA[tid][8] = 1.0f + oc * (uz * uz - uu);
    }
    __syncthreads();

    const int b = blockIdx.x * blockDim.x + tid;
    if (b >= batch) return;

    // ---- Forward chain, recording joint axes z_i and joint positions p_i ----
    float R[9] = {1.f, 0.f, 0.f, 0.f, 1.f, 0.f, 0.f, 0.f, 1.f};
    float p[3] = {0.f, 0.f, 0.f};
    float zax[DOF][3];
    float pj[DOF][3];

#pragma unroll
    for (int i = 0; i < DOF; ++i) {
        const float tx = sRaw[i * 3 + 0], ty = sRaw[i * 3 + 1], tz = sRaw[i * 3 + 2];
        p[0] += R[0] * tx + R[1] * ty + R[2] * tz;
        p[1] += R[3] * tx + R[4] * ty + R[5] * tz;
        p[2] += R[6] * tx + R[7] * ty + R[8] * tz;

        // M = R @ A_i
        float M[9];
#pragma unroll
        for (int r = 0; r < 3; ++r) {
            M[r * 3 + 0] = R[r * 3 + 0] * sA[i][0] + R[r * 3 + 1] * sA[i][3] + R[r * 3 + 2] * sA[i][6];
            M[r * 3 + 1] = R[r * 3 + 0] * sA[i][1] + R[r * 3 + 1] * sA[i][4] + R[r * 3 + 2] * sA[i][7];
            M[r * 3 + 2] = R[r * 3 + 0] * sA[i][2] + R[r * 3 + 1] * sA[i][5] + R[r * 3 + 2] * sA[i][8];
        }

        pj[i][0] = p[0]; pj[i][1] = p[1]; pj[i][2] = p[2];
        zax[i][0] = M[2]; zax[i][1] = M[5]; zax[i][2] = M[8];  // 3rd column (Rz-invariant)

        // R = M @ Rz(theta_i): only columns 0,1 change
        const float th = thetas[(size_t)b * DOF + i];
        const float s = sinf(th), c = cosf(th);
#pragma unroll
        for (int r = 0; r < 3; ++r) {
            const float m0 = M[r * 3 + 0], m1 = M[r * 3 + 1];
            R[r * 3 + 0] =  m0 * c + m1 * s;
            R[r * 3 + 1] = -m0 * s + m1 * c;
            R[r * 3 + 2] =  M[r * 3 + 2];
        }
    }

    // ---- gamma_rot = log(R) (same guarded form as reference) ----
    const float tr = R[0] + R[4] + R[8];
    const float cc = fminf(1.0f, fmaxf(-1.0f, (tr - 1.0f) * 0.5f));
    const float ang = acosf(cc);
    const float w0 = R[7] - R[5], w1 = R[2] - R[6], w2 = R[3] - R[1];
    const float sa = sinf(ang);
    const bool  safe = (sa > 1e-5f);
    const float factor = safe ? (ang / (2.0f * sa)) : 0.5f;
    const float om[3] = {factor * w0, factor * w1, factor * w2};

    // inverse left Jacobian coefficient: Jl^{-1} = I - 0.5*om^ + kc*om^2 ; |om| == ang
    const float kc = (safe && ang > 1e-4f)
                         ? (1.0f / (ang * ang) - (1.0f + cc) / (2.0f * ang * sa))
                         : (1.0f / 12.0f);

    // ---- Assemble [6 x 8] output block: J columns 0..6, gamma in column 7 ----
    float res[48];
#pragma unroll
    for (int i = 0; i < DOF; ++i) {
        const float d[3] = {p[0] - pj[i][0], p[1] - pj[i][1], p[2] - pj[i][2]};
        float jp[3], c1[3], c2[3];
        cross3(zax[i], d, jp);      // dp/dtheta_i = z_i x (p - p_i)
        cross3(om, zax[i], c1);
        cross3(om, c1, c2);
#pragma unroll
        for (int r = 0; r < 3; ++r) {
            res[r * 8 + i]       = jp[r];
            res[(r + 3) * 8 + i] = zax[i][r] - 0.5f * c1[r] + kc * c2[r];
        }
    }
#pragma unroll
    for (int r = 0; r < 3; ++r) {
        res[r * 8 + 7]       = p[r];
        res[(r + 3) * 8 + 7] = om[r];
    }

    // ---- 12 aligned 16B stores (global_store_b128), dense 192B per item ----
    float4* o4 = reinterpret_cast<float4*>(out + (size_t)b * 48);
#pragma unroll
    for (int k = 0; k < 12; ++k) {
        o4[k] = make_float4(res[4 * k + 0], res[4 * k + 1], res[4 * k + 2], res[4 * k + 3]);
    }
}

extern "C" void kernel_launch(void* const* d_in, const int* in_sizes, int n_in,
                              void* d_out, int out_size, void* d_ws, size_t ws_size,
                              hipStream_t stream) {
    (void)n_in; (void)out_size; (void)d_ws; (void)ws_size;
    const float* thetas      = (const float*)d_in[0];
    const float* link_trans  = (const float*)d_in[1];
    const float* link_rotvec = (const float*)d_in[2];
    float* out = (float*)d_out;

    const int batch = in_sizes[0] / DOF;
    const int block = 256;                 // 8 wave32 waves per workgroup
    const int grid  = (batch + block - 1) / block;
    fk_jac_kernel<<<grid, block, 0, stream>>>(thetas, link_trans, link_rotvec, out, batch);
}